// CTCLayer_80229989089978
// MI455X (gfx1250) — compile-verified
//
#include <hip/hip_runtime.h>
#include <stdint.h>

namespace {
constexpr int kB = 256;
constexpr int kT = 1024;
constexpr int kC = 96;
constexpr int kL = 64;
constexpr int kS = 2 * kL + 1;   // 129 extended states
constexpr int kSPad = 160;       // 5 states per lane * 32 lanes (states >=129 are inert padding)
constexpr int kBlank = kC - 1;   // 95
constexpr int kDepth = 8;        // async prefetch pipeline depth (rows in flight)
constexpr float kEps = 1e-7f;
constexpr float kNeg = -1e30f;
}  // namespace

#if defined(__has_builtin)
#if __has_builtin(__builtin_amdgcn_global_load_async_to_lds_b32)
#define CTC_HAVE_ASYNC_BUILTIN 1
#endif
#if __has_builtin(__builtin_amdgcn_s_wait_asynccnt)
#define CTC_HAVE_WAITASYNC_BUILTIN 1
#endif
#endif

// One lane-contiguous 128B chunk: 32 lanes x b32, global -> LDS, tracked on ASYNCcnt.
__device__ __forceinline__ void async_load_f32(const float* gsrc, float* ldst) {
#if defined(CTC_HAVE_ASYNC_BUILTIN)
  typedef __attribute__((address_space(1))) int GlobalInt;
  typedef __attribute__((address_space(3))) int LdsInt;
  __builtin_amdgcn_global_load_async_to_lds_b32(
      (GlobalInt*)(uintptr_t)gsrc,
      (LdsInt*)(uint32_t)(uintptr_t)ldst,  // low 32 bits of a flat LDS pointer == LDS offset
      0, 0);
#else
  asm volatile("global_load_async_to_lds_b32 %0, %1, off"
               :
               : "v"((uint32_t)(uintptr_t)ldst), "v"(gsrc)
               : "memory");
#endif
}

// One 384B probability row = 3 x (32 lanes x b32); global/LDS chunk deltas match (+128B).
__device__ __forceinline__ void async_row3(const float* g, float* l) {
  async_load_f32(g, l);
  async_load_f32(g + 32, l + 32);
  async_load_f32(g + 64, l + 64);
}

template <int N>
__device__ __forceinline__ void wait_asynccnt() {
#if defined(CTC_HAVE_WAITASYNC_BUILTIN)
  __builtin_amdgcn_s_wait_asynccnt(N);
#else
  asm volatile("s_wait_asynccnt %0" : : "n"(N));
#endif
}

// One alpha-recursion step reading probability row from LDS ring buffer BUF after
// the ASYNCcnt has drained to WAITN (i.e. that row is resident).
template <int BUF, int WAITN>
__device__ __forceinline__ void ctc_step(const int lane, float (&av)[5],
                                         const int (&cls)[5], const bool (&skip)[5],
                                         float* sAlpha, const float (*sRowAll)[kC]) {
  wait_asynccnt<WAITN>();
  asm volatile("" ::: "memory");
  const float* row = sRowAll[BUF];
  float nv[5];
#pragma unroll
  for (int k = 0; k < 5; ++k) {
    const int s = lane + 32 * k;
    const float a = av[k];  // own alpha kept in registers
    float a1, a2;
    if (k == 0) {  // only s = lane can hit the s-1 / s-2 boundary
      const float a1l = sAlpha[(s >= 1) ? s - 1 : 0];
      a1 = (s >= 1) ? a1l : kNeg;
      const float a2l = sAlpha[(s >= 2) ? s - 2 : 0];
      a2 = skip[0] ? a2l : kNeg;
    } else {
      a1 = sAlpha[s - 1];
      a2 = skip[k] ? sAlpha[s - 2] : kNeg;
    }
    const float lp = __logf(row[cls[k]] + kEps);
    const float m = fmaxf(a, fmaxf(a1, a2));
    const float sum = __expf(a - m) + __expf(a1 - m) + __expf(a2 - m);
    nv[k] = m + __logf(sum) + lp;
  }
  // All lanes' alpha reads are issued (and consumed) before any alpha writes;
  // same-wave in-order LDS makes the in-place update race-free.
  __builtin_amdgcn_wave_barrier();
  asm volatile("" ::: "memory");
#pragma unroll
  for (int k = 0; k < 5; ++k) {
    sAlpha[lane + 32 * k] = nv[k];
    av[k] = nv[k];
  }
  __builtin_amdgcn_wave_barrier();
  asm volatile("" ::: "memory");
}

__global__ __launch_bounds__(32) void ctc_loss_kernel(const int* __restrict__ labels,
                                                      const float* __restrict__ probs,
                                                      float* __restrict__ loss) {
  __shared__ float sAlpha[kSPad];
  __shared__ float sRow[kDepth][kC];

  const int b = blockIdx.x;
  const int lane = threadIdx.x;
  const float* rowBase = probs + (size_t)b * kT * kC;
  const int* lab = labels + b * kL;

  // Per-lane metadata for states s = lane + 32k: class index + skip-transition validity.
  // ext: even s -> blank, odd s=2i+1 -> lab[i]. Labels are in [0, C-2], never blank.
  // skip_ok: odd s>=3 and lab[i]!=lab[i-1]; s==1 padded-NEG in the reference => false here.
  int cls[5];
  bool skip[5];
#pragma unroll
  for (int k = 0; k < 5; ++k) {
    const int s = lane + 32 * k;
    int c = kBlank;
    bool sk = false;
    if (s < kS && (s & 1)) {
      const int i = (s - 1) >> 1;
      c = lab[i];
      if (s >= 3) sk = (c != lab[i - 1]);
    }
    cls[k] = c;
    skip[k] = sk;
  }

  // Prologue: fill the async pipeline with rows 0..7.
#pragma unroll
  for (int d = 0; d < kDepth; ++d) {
    async_row3(rowBase + d * kC + lane, &sRow[d][lane]);
  }
  wait_asynccnt<3 * (kDepth - 1)>();  // row 0 resident in LDS
  asm volatile("" ::: "memory");

  // t = 0: alpha0[0] = lp(blank), alpha0[1] = lp(lab[0]), else NEG.
  float av[5];
#pragma unroll
  for (int k = 0; k < 5; ++k) {
    const int s = lane + 32 * k;
    const float p = sRow[0][cls[k]];
    const float a = (s <= 1) ? __logf(p + kEps) : kNeg;
    av[k] = a;
    sAlpha[s] = a;
  }
  __builtin_amdgcn_wave_barrier();
  asm volatile("" ::: "memory");

  // Main loop: t = 1..1016 in blocks of 8 so every ring-buffer index is a
  // compile-time constant. Slot j (t = 1+8i+j) prefetches row t+7 into buf j
  // (row (t+7)%8 == j) and consumes buf (j+1)&7 (row t%8 == (j+1)&7).
  const float* gpref = rowBase + (size_t)kDepth * kC + lane;  // next row to prefetch (row 8)
#define CTC_SLOT(J)                                                           \
  async_row3(gpref, &sRow[(J)][lane]);                                        \
  gpref += kC;                                                                \
  ctc_step<((J) + 1) & 7, 3 * (kDepth - 1)>(lane, av, cls, skip, sAlpha, sRow);

  for (int i = 0; i < (kT - kDepth) / kDepth; ++i) {  // 127 blocks of 8 steps
    CTC_SLOT(0) CTC_SLOT(1) CTC_SLOT(2) CTC_SLOT(3)
    CTC_SLOT(4) CTC_SLOT(5) CTC_SLOT(6) CTC_SLOT(7)
  }
#undef CTC_SLOT

  // Epilogue: t = 1017..1023. No prefetches remain; drain the pipeline with
  // per-step constant waits (rows t..1023 outstanding => wait 3*(1023-t)).
  ctc_step<1, 18>(lane, av, cls, skip, sAlpha, sRow);
  ctc_step<2, 15>(lane, av, cls, skip, sAlpha, sRow);
  ctc_step<3, 12>(lane, av, cls, skip, sAlpha, sRow);
  ctc_step<4, 9>(lane, av, cls, skip, sAlpha, sRow);
  ctc_step<5, 6>(lane, av, cls, skip, sAlpha, sRow);
  ctc_step<6, 3>(lane, av, cls, skip, sAlpha, sRow);
  ctc_step<7, 0>(lane, av, cls, skip, sAlpha, sRow);

  if (lane == 0) {
    const float aN1 = sAlpha[kS - 1];
    const float aN2 = sAlpha[kS - 2];
    const float m = fmaxf(aN1, aN2);
    loss[b] = -(m + __logf(__expf(aN1 - m) + __expf(aN2 - m)));
  }
}

extern "C" void kernel_launch(void* const* d_in, const int* in_sizes, int n_in,
                              void* d_out, int out_size, void* d_ws, size_t ws_size,
                              hipStream_t stream) {
  (void)in_sizes; (void)n_in; (void)d_ws; (void)ws_size; (void)out_size;
  const int* y_true = (const int*)d_in[0];     // [B, L] int32
  const float* y_pred = (const float*)d_in[1]; // [B, T, C] float32
  float* out = (float*)d_out;                  // [B, 1] float32
  ctc_loss_kernel<<<dim3(kB), dim3(32), 0, stream>>>(y_true, y_pred, out);
}